// ConvNeXTBlock_12446815224088
// MI455X (gfx1250) — compile-verified
//
#include <hip/hip_runtime.h>

typedef __attribute__((ext_vector_type(2))) float v2f;
typedef __attribute__((ext_vector_type(8))) float v8f;

#define NTOK   131072          // 32*64*64 tokens
#define CCH    128
#define DBOT   512
#define HWIMG  4096            // 64*64
#define HPAD   516             // 512+4: 516%64==4 -> conflict-free LDS
#define OPAD   132             // 128+4

// -------- Kernel 1: depthwise 7x7 conv + bias + channel LayerNorm ----------
// One wave32 per pixel; lane owns channels {lane, lane+32, lane+64, lane+96}.
__global__ __launch_bounds__(256) void dwconv_ln_kernel(
    const float* __restrict__ x, const float* __restrict__ dw_w,
    const float* __restrict__ dw_b, const float* __restrict__ gamma,
    const float* __restrict__ beta, float* __restrict__ yln)
{
    const int tid  = threadIdx.x;
    const int lane = tid & 31;
    const int wave = tid >> 5;
    const int p  = blockIdx.x * 8 + wave;     // token index
    const int n  = p >> 12;
    const int hw = p & (HWIMG - 1);
    const int h  = hw >> 6;
    const int w  = hw & 63;

    float acc[4];
#pragma unroll
    for (int j = 0; j < 4; ++j) {
        const int c = lane + 32 * j;
        float a = dw_b[c];
        const float* xc = x + (size_t)(n * CCH + c) * HWIMG;
        const float* wc = dw_w + c * 49;
#pragma unroll
        for (int ky = 0; ky < 7; ++ky) {
            const int yy = h + ky - 3;
            if (yy < 0 || yy > 63) continue;
#pragma unroll
            for (int kx = 0; kx < 7; ++kx) {
                const int xx = w + kx - 3;
                if (xx < 0 || xx > 63) continue;
                a += xc[yy * 64 + xx] * wc[ky * 7 + kx];
            }
        }
        acc[j] = a;
    }

    float s  = acc[0] + acc[1] + acc[2] + acc[3];
    float s2 = acc[0]*acc[0] + acc[1]*acc[1] + acc[2]*acc[2] + acc[3]*acc[3];
#pragma unroll
    for (int m = 16; m >= 1; m >>= 1) {        // wave32 reduction
        s  += __shfl_xor(s,  m, 32);
        s2 += __shfl_xor(s2, m, 32);
    }
    const float mean = s * (1.0f / 128.0f);
    const float var  = s2 * (1.0f / 128.0f) - mean * mean;
    const float rs   = rsqrtf(var + 1e-5f);

#pragma unroll
    for (int j = 0; j < 4; ++j) {
        const int c = lane + 32 * j;
        yln[(size_t)p * CCH + c] = (acc[j] - mean) * rs * gamma[c] + beta[c];
    }
}

// -------- Kernel 2: pw1 (relu) + pw2 + residual, WMMA f32 16x16x4 ----------
// 256 threads = 8 waves per 16-token tile. pw1: wave q owns d-tiles q*4..q*4+3.
// pw2: wave q owns c-tile q. H staged in LDS [16][HPAD], O in LDS [16][OPAD].
__global__ __launch_bounds__(256) void mlp_kernel(
    const float* __restrict__ x,  const float* __restrict__ yln,
    const float* __restrict__ w1, const float* __restrict__ b1,
    const float* __restrict__ w2, const float* __restrict__ b2,
    float* __restrict__ out)
{
    __shared__ float Hs[16 * HPAD];
    __shared__ float Os[16 * OPAD];

    const int tid  = threadIdx.x;
    const int lane = tid & 31;
    const int q    = tid >> 5;                 // wave 0..7
    const int t0   = blockIdx.x * 16;          // first token of tile
    const int lm   = lane & 15;                // N / M index within tile
    const int kk   = (lane >> 4) << 1;         // K sub-offset: 0 or 2
    const int rofs = (lane & 16) ? 8 : 0;      // C/D row offset for half-wave

    // ---------------- pw1: H = relu(Y * W1^T + b1) ----------------
    v8f zero = {};
    v8f acc[4];
#pragma unroll
    for (int j = 0; j < 4; ++j) acc[j] = zero;

    const float* arow = yln + (size_t)(t0 + lm) * CCH + kk;   // A: M=lm, K=k+kk
    for (int k = 0; k < CCH; k += 4) {
        const v2f a = *(const v2f*)(arow + k);
#pragma unroll
        for (int j = 0; j < 4; ++j) {
            const int d = (q * 4 + j) * 16 + lm;              // B: N=d, K=k+kk
            const v2f b = *(const v2f*)(w1 + d * CCH + k + kk);
            acc[j] = __builtin_amdgcn_wmma_f32_16x16x4_f32(
                false, a, false, b, (short)0, acc[j], false, false);
        }
    }

#pragma unroll
    for (int j = 0; j < 4; ++j) {
        const int d = (q * 4 + j) * 16 + lm;
        const float bias = b1[d];
#pragma unroll
        for (int r = 0; r < 8; ++r) {
            float v = acc[j][r] + bias;
            v = v > 0.0f ? v : 0.0f;                          // relu
            Hs[(r + rofs) * HPAD + d] = v;
        }
    }
    __syncthreads();

    // ---------------- pw2: O = H * W2^T + b2 ----------------
    v8f oacc = zero;
    const float* hrow = Hs + lm * HPAD + kk;                  // A from LDS
    const int c = q * 16 + lm;                                // B column
    const float* brow = w2 + c * DBOT + kk;
    for (int k = 0; k < DBOT; k += 4) {
        const v2f a = *(const v2f*)(hrow + k);
        const v2f b = *(const v2f*)(brow + k);
        oacc = __builtin_amdgcn_wmma_f32_16x16x4_f32(
            false, a, false, b, (short)0, oacc, false, false);
    }
    {
        const float bias = b2[c];
#pragma unroll
        for (int r = 0; r < 8; ++r)
            Os[(r + rofs) * OPAD + c] = oacc[r] + bias;
    }
    __syncthreads();

    // ---------------- residual add + NCHW store ----------------
    const int n   = t0 >> 12;
    const int hw0 = t0 & (HWIMG - 1);
#pragma unroll
    for (int i = 0; i < 8; ++i) {
        const int idx = i * 256 + tid;          // 0..2047
        const int cc  = idx >> 4;               // channel 0..127
        const int mm  = idx & 15;               // token-in-tile
        const size_t g = (size_t)(n * CCH + cc) * HWIMG + hw0 + mm;
        out[g] = x[g] + Os[mm * OPAD + cc];
    }
}

extern "C" void kernel_launch(void* const* d_in, const int* in_sizes, int n_in,
                              void* d_out, int out_size, void* d_ws, size_t ws_size,
                              hipStream_t stream) {
    const float* x     = (const float*)d_in[0];
    const float* dw_w  = (const float*)d_in[1];
    const float* dw_b  = (const float*)d_in[2];
    const float* gamma = (const float*)d_in[3];
    const float* beta  = (const float*)d_in[4];
    const float* pw1_w = (const float*)d_in[5];
    const float* pw1_b = (const float*)d_in[6];
    const float* pw2_w = (const float*)d_in[7];
    const float* pw2_b = (const float*)d_in[8];
    float* out = (float*)d_out;
    float* yln = (float*)d_ws;                  // 131072*128*4 = 67 MB scratch

    dwconv_ln_kernel<<<NTOK / 8, 256, 0, stream>>>(x, dw_w, dw_b, gamma, beta, yln);
    mlp_kernel<<<NTOK / 16, 256, 0, stream>>>(x, yln, pw1_w, pw1_b, pw2_w, pw2_b, out);
}